// LaplaceProductKernel_64209761075383
// MI455X (gfx1250) — compile-verified
//
#include <hip/hip_runtime.h>
#include <hip/hip_bf16.h>
#include <stdint.h>

// ---------------------------------------------------------------------------
// Laplace (L1 / exp) kernel matrix, n=m=8192, d=64, fp32.
// VALU-bound problem (L1 distance is not a matmul -> WMMA inapplicable);
// CDNA5 path used: Tensor Data Mover (tensor_load_to_lds, 6-arg clang-23
// form) with descriptor padding for conflict-free LDS, s_wait_tensorcnt,
// wave32 tiling.
// ---------------------------------------------------------------------------

#define DDIM   64      // feature dimension
#define TI     128     // output tile rows (x1)
#define TJ     128     // output tile cols (x2)
#define ROWP   68      // padded LDS row stride in floats (64 + 4 via TDM pad)

typedef __attribute__((ext_vector_type(4))) unsigned int u32x4;
typedef __attribute__((ext_vector_type(8))) int          i32x8;
typedef __attribute__((ext_vector_type(4))) int          i32x4;

// ---------------- Kernel 1: column means of x1 + 1/lengthscale -------------
__global__ __launch_bounds__(64) void colstats_kernel(
    const float* __restrict__ x1, const float* __restrict__ ls,
    float* __restrict__ adj, float* __restrict__ inv, int n)
{
    const int d = threadIdx.x;            // 64 threads, one per dimension
    float s0 = 0.f, s1 = 0.f, s2 = 0.f, s3 = 0.f;
    for (int r = 0; r < n; r += 4) {      // coalesced: 64 lanes read one row
        s0 += x1[(size_t)(r + 0) * DDIM + d];
        s1 += x1[(size_t)(r + 1) * DDIM + d];
        s2 += x1[(size_t)(r + 2) * DDIM + d];
        s3 += x1[(size_t)(r + 3) * DDIM + d];
    }
    adj[d] = (s0 + s1 + s2 + s3) / (float)n;
    inv[d] = 1.0f / ls[d];
}

// ---------------- Kernel 2: center + scale both inputs ---------------------
__global__ __launch_bounds__(256) void normalize_kernel(
    const float* __restrict__ x1, const float* __restrict__ x2,
    const float* __restrict__ adj, const float* __restrict__ inv,
    float* __restrict__ x1n, float* __restrict__ x2n, int total)
{
    const int i = blockIdx.x * blockDim.x + threadIdx.x;
    if (i >= total) return;
    const int d = i & (DDIM - 1);
    const float a = adj[d];
    const float s = inv[d];
    x1n[i] = (x1[i] - a) * s;
    x2n[i] = (x2[i] - a) * s;
}

// ---------------- TDM descriptor builder (2D tile, fp32) -------------------
// Loads a TI(/TJ) x 64 fp32 tile, row stride 64 elems in memory, into LDS
// with 4 DWORDs of padding inserted after every 64 DWORDs stored
// (pad_interval code 5 = 64 DW, pad_amount code 3 = 4 DW) -> 68-float rows.
__device__ __forceinline__ void tdm_load_tile(
    const float* gsrc, uint32_t lds_byte_addr, int tile_rows, int tensor_rows)
{
    const uint64_t ga = (uint64_t)(uintptr_t)gsrc;

    u32x4 g0;
    g0[0] = 1u;                                   // count=1, user descriptor
    g0[1] = lds_byte_addr;                        // lds_addr (bytes)
    g0[2] = (uint32_t)(ga & 0xFFFFFFFFull);       // global_addr[31:0]
    g0[3] = (uint32_t)((ga >> 32) & 0x01FFFFFFull)// global_addr[56:32]
          | 0x80000000u;                          // type=2 ("image")

    i32x8 g1;
    g1[0] = (2 << 16)       // data_size = 4 bytes
          | (1 << 20)       // pad_enable
          | (5 << 22)       // pad_interval: 64 DWORDs between pads
          | (3 << 25);      // pad_amount: 4 DWORDs of pad
    g1[1] = (DDIM & 0xFFFF) << 16;                // tensor_dim0 = 64
    g1[2] = (tensor_rows & 0xFFFF) << 16;         // tensor_dim1 low16
    g1[3] = ((tensor_rows >> 16) & 0xFFFF)        // tensor_dim1 high16
          | (DDIM << 16);                         // tile_dim0 = 64
    g1[4] = tile_rows & 0xFFFF;                   // tile_dim1, tile_dim2=0
    g1[5] = DDIM;                                 // tensor_dim0_stride = 64
    g1[6] = 0;
    g1[7] = 0;

    const i32x4 z4 = {0, 0, 0, 0};                // groups 2/3 unused (2D)
    const i32x8 z8 = {0, 0, 0, 0, 0, 0, 0, 0};    // extra group (clang-23 form)

#if __has_builtin(__builtin_amdgcn_tensor_load_to_lds)
    __builtin_amdgcn_tensor_load_to_lds(g0, g1, z4, z4, z8, 0);
#endif
}

// ---------------- Kernel 3: 128x128 Laplace tile ---------------------------
__global__ __launch_bounds__(256) void laplace_tile_kernel(
    const float* __restrict__ x1n, const float* __restrict__ x2n,
    float* __restrict__ out, int n, int m)
{
    __shared__ float sA[TI * ROWP];   // x1 tile, padded rows
    __shared__ float sB[TJ * ROWP];   // x2 tile, padded rows

    const int tid = threadIdx.x;
    const int tx  = tid & 15;         // 16 j-groups
    const int ty  = tid >> 4;         // 16 i-groups
    const int i0  = blockIdx.y * TI;
    const int j0  = blockIdx.x * TJ;

    // One real store into each shared array so LLVM cannot fold the LDS
    // loads below to undef (the TDM write is invisible to alias analysis).
    if (tid == 0) { sA[0] = 0.0f; sB[0] = 0.0f; }
    __syncthreads();

    // Wave 0 issues both tile DMAs via the Tensor Data Mover and waits on
    // its TENSORcnt; the workgroup barrier then publishes the LDS data.
    if (tid < 32) {
        const uint32_t ldsA = (uint32_t)(uintptr_t)(void*)&sA[0];
        const uint32_t ldsB = (uint32_t)(uintptr_t)(void*)&sB[0];
        tdm_load_tile(x1n + (size_t)i0 * DDIM, ldsA, TI, n);
        tdm_load_tile(x2n + (size_t)j0 * DDIM, ldsB, TJ, m);
        __builtin_amdgcn_s_wait_tensorcnt(0);
    }
    __syncthreads();

    float acc[8][8];
#pragma unroll
    for (int ii = 0; ii < 8; ++ii)
#pragma unroll
        for (int jj = 0; jj < 8; ++jj) acc[ii][jj] = 0.0f;

    // d processed in float4 chunks: 16 ds_load_b128 feed 512 VALU ops.
    // Row stride 68 DW => consecutive rows shift 4 banks: the 16 lanes
    // (rows tx+16*jj) cover all 64 banks -> conflict-free b128 reads.
#pragma unroll 4
    for (int dq = 0; dq < DDIM / 4; ++dq) {
        float4 av[8], bv[8];
#pragma unroll
        for (int u = 0; u < 8; ++u) {
            av[u] = *(const float4*)(sA + (ty + 16 * u) * ROWP + dq * 4);
            bv[u] = *(const float4*)(sB + (tx + 16 * u) * ROWP + dq * 4);
        }
#pragma unroll
        for (int ii = 0; ii < 8; ++ii) {
#pragma unroll
            for (int jj = 0; jj < 8; ++jj) {
                acc[ii][jj] += fabsf(av[ii].x - bv[jj].x)
                             + fabsf(av[ii].y - bv[jj].y)
                             + fabsf(av[ii].z - bv[jj].z)
                             + fabsf(av[ii].w - bv[jj].w);
            }
        }
    }

    // Epilogue: clamp, exp(-d) (v_exp_f32), coalesced b32 stores.
#pragma unroll
    for (int ii = 0; ii < 8; ++ii) {
        const size_t row = (size_t)(i0 + ty + 16 * ii);
        float* orow = out + row * (size_t)m + j0;
#pragma unroll
        for (int jj = 0; jj < 8; ++jj) {
            const float dd = fmaxf(acc[ii][jj], 1e-15f);
            orow[tx + 16 * jj] = __expf(-dd);
        }
    }
}

// ---------------------------------------------------------------------------
extern "C" void kernel_launch(void* const* d_in, const int* in_sizes, int n_in,
                              void* d_out, int out_size, void* d_ws, size_t ws_size,
                              hipStream_t stream)
{
    const float* x1 = (const float*)d_in[0];
    const float* x2 = (const float*)d_in[1];
    const float* ls = (const float*)d_in[2];
    float* out = (float*)d_out;

    const int n = in_sizes[0] / DDIM;   // 8192
    const int m = in_sizes[1] / DDIM;   // 8192

    float* ws   = (float*)d_ws;
    float* adj  = ws;                         // 64 floats
    float* inv  = ws + 64;                    // 64 floats
    float* x1n  = ws + 128;                   // n*64 floats (512B aligned)
    float* x2n  = x1n + (size_t)n * DDIM;     // m*64 floats

    colstats_kernel<<<1, 64, 0, stream>>>(x1, ls, adj, inv, n);

    const int total = n * DDIM;               // both arrays same size here
    normalize_kernel<<<(total + 255) / 256, 256, 0, stream>>>(
        x1, x2, adj, inv, x1n, x2n, total);

    dim3 grid(m / TJ, n / TI);                // 64 x 64 blocks
    laplace_tile_kernel<<<grid, 256, 0, stream>>>(x1n, x2n, out, n, m);
}